// SSDDecoder_62027917689560
// MI455X (gfx1250) — compile-verified
//
#include <hip/hip_runtime.h>
#include <stdint.h>

#define BATCH   64
#define NANCH   8732
#define NCLS    21
#define KCAND   200
#define PMAX    20
#define MAXTOT  20
#define SCORE_TH 0.5f
#define IOU_TH   0.5f
#define TPB     256
#define FTPB    512

typedef unsigned int v4u __attribute__((ext_vector_type(4)));
typedef int          v8i __attribute__((ext_vector_type(8)));
typedef int          v4i __attribute__((ext_vector_type(4)));

// ---------------------------------------------------------------------------
// K1: decode anchors+deltas -> boxes; per-anchor background mask (argmax != 0)
// ---------------------------------------------------------------------------
__global__ __launch_bounds__(TPB) void k_decode_mask(
    const float* __restrict__ deltas, const float* __restrict__ labels,
    const float* __restrict__ anchors, float* __restrict__ boxes,
    float* __restrict__ mask)
{
    int i = blockIdx.x * blockDim.x + threadIdx.x;
    if (i >= BATCH * NANCH) return;
    int n = i % NANCH;

    float a0 = anchors[n * 4 + 0], a1 = anchors[n * 4 + 1];
    float a2 = anchors[n * 4 + 2], a3 = anchors[n * 4 + 3];
    float ah = a2 - a0, aw = a3 - a1;
    float acy = a0 + 0.5f * ah, acx = a1 + 0.5f * aw;

    const float* d = deltas + (size_t)i * 4;
    float d0 = d[0] * 0.1f, d1 = d[1] * 0.1f;
    float d2 = d[2] * 0.2f, d3 = d[3] * 0.2f;
    float cy = d0 * ah + acy, cx = d1 * aw + acx;
    float h  = expf(d2) * ah, w = expf(d3) * aw;

    float* ob = boxes + (size_t)i * 4;
    ob[0] = cy - 0.5f * h; ob[1] = cx - 0.5f * w;
    ob[2] = cy + 0.5f * h; ob[3] = cx + 0.5f * w;

    const float* L = labels + (size_t)i * NCLS;
    float best = L[0]; int bi = 0;
    #pragma unroll
    for (int c = 1; c < NCLS; ++c) { float v = L[c]; if (v > best) { best = v; bi = c; } }
    mask[i] = (bi != 0) ? 1.0f : 0.0f;
}

// ---------------------------------------------------------------------------
// CDNA5 async global->LDS helpers (ASYNCcnt path)
// ---------------------------------------------------------------------------
__device__ __forceinline__ void async_load_b32_to_lds(uint32_t lds_byte, uint64_t gaddr)
{
    asm volatile("global_load_async_to_lds_b32 %0, %1, off"
                 :: "v"(lds_byte), "v"(gaddr) : "memory");
}
__device__ __forceinline__ void async_load_b128_to_lds(uint32_t lds_byte, uint64_t gaddr)
{
    asm volatile("global_load_async_to_lds_b128 %0, %1, off"
                 :: "v"(lds_byte), "v"(gaddr) : "memory");
}
__device__ __forceinline__ void wait_async0()
{
    asm volatile("s_wait_asynccnt 0" ::: "memory");
}
__device__ __forceinline__ void wait_tensor0()
{
#if __has_builtin(__builtin_amdgcn_s_wait_tensorcnt)
    __builtin_amdgcn_s_wait_tensorcnt(0);
#else
    asm volatile("s_wait_tensorcnt 0" ::: "memory");
#endif
}

// ---------------------------------------------------------------------------
// K2: per (batch, class) exact top-200 via 64-bit composite radix-select.
//     The 84-byte-strided class column is gathered into LDS by the Tensor
//     Data Mover (one 2D descriptor, TENSORcnt) -- fallback: async loads.
// ---------------------------------------------------------------------------
__global__ __launch_bounds__(TPB) void k_topk(
    const float* __restrict__ labels, const float* __restrict__ mask,
    const float* __restrict__ boxes,
    float* __restrict__ cvals, float* __restrict__ cboxes)
{
    __shared__ uint32_t           skey[NANCH];   // masked score bits per anchor
    __shared__ unsigned           hist[256];
    __shared__ unsigned long long cand[256];
    __shared__ unsigned long long s_prefix;
    __shared__ unsigned           s_remaining;
    __shared__ unsigned           s_counter;

    const int tid = threadIdx.x;
    const int b   = blockIdx.x / NCLS;
    const int c   = blockIdx.x % NCLS;
    const float* Lb = labels + (size_t)b * NANCH * NCLS + c;

    const uint32_t lds0 = (uint32_t)(uintptr_t)&skey[0];
    const uint64_t ga   = (uint64_t)(uintptr_t)Lb;

#if __has_builtin(__builtin_amdgcn_tensor_load_to_lds)
    // ---- TDM: gather the strided column (stride = NCLS*4 B) into LDS -----
    if (tid < 32) {                       // single wave programs the DMA
        // D# group0: count=1 | lds_addr | 57b global_addr | type=2
        v4u g0;
        g0[0] = 1u;                                              // count=1, user mode
        g0[1] = lds0;                                            // lds_addr
        g0[2] = (uint32_t)(ga & 0xFFFFFFFFu);                    // global_addr[31:0]
        g0[3] = (uint32_t)((ga >> 32) & 0x1FFFFFFu) | (2u << 30);// ga[56:32] | type=2
        // D# group1: data_size=4B, tensor_dim0=1, tensor_dim1=N,
        //            tile_dim0=1, tile_dim1=N, dim0_stride=NCLS
        v8i g1;
        g1[0] = (int)(2u << 16);                                 // data_size=2 (4B)
        g1[1] = (int)(1u << 16);                                 // tensor_dim0[15:0]=1
        g1[2] = (int)(((uint32_t)NANCH & 0xFFFFu) << 16);        // tensor_dim1[15:0]=N
        g1[3] = (int)(1u << 16);                                 // tile_dim0=1
        g1[4] = (int)((uint32_t)NANCH & 0xFFFFu);                // tile_dim1=N, tile_dim2=0
        g1[5] = (int)NCLS;                                       // tensor_dim0_stride=21
        g1[6] = 0;
        g1[7] = 0;
        v4i g2 = {0, 0, 0, 0};                                   // 2D tile: unused
        v4i g3 = {0, 0, 0, 0};                                   // 2D tile: unused
        v8i g4 = {0, 0, 0, 0, 0, 0, 0, 0};                       // 6-arg form extra group
        __builtin_amdgcn_tensor_load_to_lds(g0, g1, g2, g3, g4, 0);
        wait_tensor0();
    }
#else
    // ---- fallback: per-lane async loads (ASYNCcnt) ------------------------
    for (int n = tid; n < NANCH; n += TPB) {
        async_load_b32_to_lds(lds0 + (uint32_t)n * 4u,
                              (uint64_t)(uintptr_t)(Lb + (size_t)n * NCLS));
    }
    wait_async0();
#endif
    __syncthreads();

    // ---- apply background mask in LDS (scores >= 0 so raw bits are ordered)
    const float* mrow = mask + (size_t)b * NANCH;
    for (int n = tid; n < NANCH; n += TPB) {
        uint32_t kb = skey[n];
        skey[n] = (mrow[n] != 0.0f) ? kb : 0u;
    }
    __syncthreads();

    // ---- radix-select the 200th-largest composite key (score<<32 | ~idx) --
    unsigned long long prefix = 0ull;
    unsigned remaining = KCAND;
    for (int pass = 0; pass < 8; ++pass) {
        int shift = 56 - pass * 8;
        unsigned long long pmask = (pass == 0) ? 0ull : (~0ull << (shift + 8));
        hist[tid] = 0u;
        __syncthreads();
        for (int n = tid; n < NANCH; n += TPB) {
            unsigned long long kk =
                ((unsigned long long)skey[n] << 32) | (0xFFFFFFFFu - (unsigned)n);
            if ((kk & pmask) == prefix)
                atomicAdd(&hist[(unsigned)((kk >> shift) & 0xFF)], 1u);
        }
        __syncthreads();
        if (tid == 0) {
            unsigned sum = 0; int chosen = 0;
            for (int bin = 255; bin >= 0; --bin) {
                unsigned h = hist[bin];
                if (sum + h >= remaining) { chosen = bin; break; }
                sum += h;
            }
            s_prefix    = prefix | ((unsigned long long)(unsigned)chosen << shift);
            s_remaining = remaining - sum;
            s_counter   = 0u;
        }
        __syncthreads();
        prefix    = s_prefix;
        remaining = s_remaining;
        __syncthreads();
    }
    const unsigned long long T = prefix;   // exact 200th-largest key (keys distinct)

    // ---- collect exactly the 200 keys >= T --------------------------------
    cand[tid] = ~0ull;
    __syncthreads();
    for (int n = tid; n < NANCH; n += TPB) {
        unsigned long long kk =
            ((unsigned long long)skey[n] << 32) | (0xFFFFFFFFu - (unsigned)n);
        if (kk >= T) {
            unsigned p = atomicAdd(&s_counter, 1u);
            if (p < KCAND) cand[p] = ~kk;  // invert: ascending sort == descending rank
        }
    }
    __syncthreads();

    // ---- bitonic sort 256 keys (ascending => descending score, stable idx)
    for (unsigned sz = 2; sz <= 256; sz <<= 1) {
        for (unsigned j = sz >> 1; j > 0; j >>= 1) {
            __syncthreads();
            unsigned t = (unsigned)tid, ixj = t ^ j;
            if (ixj > t) {
                unsigned long long A = cand[t], B = cand[ixj];
                bool up = ((t & sz) == 0);
                if ((A > B) == up) { cand[t] = B; cand[ixj] = A; }
            }
        }
    }
    __syncthreads();

    // ---- emit sorted candidate scores + gathered boxes --------------------
    if (tid < KCAND) {
        unsigned long long kk = ~cand[tid];
        uint32_t vb = (uint32_t)(kk >> 32);
        unsigned n  = 0xFFFFFFFFu - (uint32_t)(kk & 0xFFFFFFFFu);
        float val = __uint_as_float(vb);
        size_t o = (size_t)blockIdx.x * KCAND + tid;
        cvals[o] = val;
        float b0 = 0.f, b1 = 0.f, b2 = 0.f, b3 = 0.f;
        if (n < NANCH) {
            const float* bp = boxes + ((size_t)b * NANCH + n) * 4;
            b0 = bp[0]; b1 = bp[1]; b2 = bp[2]; b3 = bp[3];
        }
        cboxes[o * 4 + 0] = b0; cboxes[o * 4 + 1] = b1;
        cboxes[o * 4 + 2] = b2; cboxes[o * 4 + 3] = b3;
    }
}

// ---------------------------------------------------------------------------
// K3: greedy NMS over 200 sorted candidates, compact first 20 kept.
//     Candidates staged into LDS via CDNA5 async b32/b128 loads.
// ---------------------------------------------------------------------------
__global__ __launch_bounds__(TPB) void k_nms(
    const float* __restrict__ cvals, const float* __restrict__ cboxes,
    float* __restrict__ kvals, float* __restrict__ kboxes)
{
    __shared__ float                 v[KCAND];
    __shared__ __align__(16) float   bx[KCAND][4];
    __shared__ int                   keep[KCAND];

    const int tid = threadIdx.x;
    const int bc  = blockIdx.x;

    // ---- async-stage scores (b32) and boxes (b128) straight into LDS ------
    const uint32_t vbase = (uint32_t)(uintptr_t)&v[0];
    const uint32_t bbase = (uint32_t)(uintptr_t)&bx[0][0];
    for (int i = tid; i < KCAND; i += TPB) {
        async_load_b32_to_lds (vbase + 4u  * (uint32_t)i,
                               (uint64_t)(uintptr_t)(cvals + (size_t)bc * KCAND + i));
        async_load_b128_to_lds(bbase + 16u * (uint32_t)i,
                               (uint64_t)(uintptr_t)(cboxes + ((size_t)bc * KCAND + i) * 4));
    }
    wait_async0();
    __syncthreads();

    const bool lane = (tid < KCAND);
    float my0 = 0.f, my1 = 0.f, my2 = 0.f, my3 = 0.f, myA = 0.f;
    bool sup = false;
    if (lane) {
        my0 = bx[tid][0]; my1 = bx[tid][1]; my2 = bx[tid][2]; my3 = bx[tid][3];
        myA = (my2 - my0) * (my3 - my1);
    }
    for (int i = 0; i < KCAND; ++i) {
        if (tid == i) keep[i] = ((v[i] > SCORE_TH) && !sup) ? 1 : 0;
        __syncthreads();
        if (keep[i] && lane && tid > i) {
            float o0 = bx[i][0], o1 = bx[i][1], o2 = bx[i][2], o3 = bx[i][3];
            float oA = (o2 - o0) * (o3 - o1);
            float ih = fminf(my2, o2) - fmaxf(my0, o0); ih = ih > 0.f ? ih : 0.f;
            float iw = fminf(my3, o3) - fmaxf(my1, o1); iw = iw > 0.f ? iw : 0.f;
            float inter = ih * iw;
            float iou = inter / (myA + oA - inter + 1e-8f);
            if (iou > IOU_TH) sup = true;
        }
        __syncthreads();
    }

    if (tid == 0) {
        int out = 0;
        size_t ob = (size_t)bc * PMAX;
        for (int i = 0; i < KCAND && out < PMAX; ++i) {
            if (keep[i]) {
                kvals[ob + out] = v[i];
                kboxes[(ob + out) * 4 + 0] = bx[i][0];
                kboxes[(ob + out) * 4 + 1] = bx[i][1];
                kboxes[(ob + out) * 4 + 2] = bx[i][2];
                kboxes[(ob + out) * 4 + 3] = bx[i][3];
                ++out;
            }
        }
        for (; out < PMAX; ++out) {
            kvals[ob + out] = 0.f;
            kboxes[(ob + out) * 4 + 0] = 0.f; kboxes[(ob + out) * 4 + 1] = 0.f;
            kboxes[(ob + out) * 4 + 2] = 0.f; kboxes[(ob + out) * 4 + 3] = 0.f;
        }
    }
}

// ---------------------------------------------------------------------------
// K4: per-batch top-20 over the 21*20=420 kept candidates; clip + label.
// ---------------------------------------------------------------------------
__global__ __launch_bounds__(FTPB) void k_final(
    const float* __restrict__ kvals, const float* __restrict__ kboxes,
    float* __restrict__ out_boxes, float* __restrict__ out_vals,
    float* __restrict__ out_labels)
{
    const int M = NCLS * PMAX; // 420
    __shared__ unsigned long long key[FTPB];
    const int b = blockIdx.x, tid = threadIdx.x;

    if (tid < M) {
        float vv = kvals[(size_t)b * M + tid];          // always >= 0
        uint32_t bits = __float_as_uint(vv);
        key[tid] = ((unsigned long long)(~bits) << 32) | (unsigned)tid;
    } else {
        key[tid] = ~0ull;                               // sorts last
    }
    for (unsigned sz = 2; sz <= FTPB; sz <<= 1) {
        for (unsigned j = sz >> 1; j > 0; j >>= 1) {
            __syncthreads();
            unsigned t = (unsigned)tid, ixj = t ^ j;
            if (ixj > t) {
                unsigned long long A = key[t], B = key[ixj];
                bool up = ((t & sz) == 0);
                if ((A > B) == up) { key[t] = B; key[ixj] = A; }
            }
        }
    }
    __syncthreads();

    if (tid < MAXTOT) {
        unsigned long long kk = key[tid];
        float val   = __uint_as_float(~(uint32_t)(kk >> 32));
        unsigned q  = (uint32_t)(kk & 0xFFFFFFFFu);
        bool valid  = val > 0.0f;
        out_vals[(size_t)b * MAXTOT + tid]   = val;
        out_labels[(size_t)b * MAXTOT + tid] = valid ? (float)(q / PMAX) : 0.0f;
        float bb[4] = {0.f, 0.f, 0.f, 0.f};
        if (valid) {
            const float* p = kboxes + ((size_t)b * M + q) * 4;
            #pragma unroll
            for (int k = 0; k < 4; ++k) bb[k] = fminf(fmaxf(p[k], 0.f), 1.f);
        }
        #pragma unroll
        for (int k = 0; k < 4; ++k)
            out_boxes[((size_t)b * MAXTOT + tid) * 4 + k] = bb[k];
    }
}

// ---------------------------------------------------------------------------
extern "C" void kernel_launch(void* const* d_in, const int* in_sizes, int n_in,
                              void* d_out, int out_size, void* d_ws, size_t ws_size,
                              hipStream_t stream)
{
    (void)in_sizes; (void)n_in; (void)out_size; (void)ws_size;
    const float* deltas  = (const float*)d_in[0];
    const float* labels  = (const float*)d_in[1];
    const float* anchors = (const float*)d_in[2];

    float* ws        = (float*)d_ws;
    float* ws_boxes  = ws;                                           // B*N*4
    float* ws_mask   = ws_boxes  + (size_t)BATCH * NANCH * 4;        // B*N
    float* ws_cvals  = ws_mask   + (size_t)BATCH * NANCH;            // B*C*K
    float* ws_cboxes = ws_cvals  + (size_t)BATCH * NCLS * KCAND;     // B*C*K*4
    float* ws_kvals  = ws_cboxes + (size_t)BATCH * NCLS * KCAND * 4; // B*C*P
    float* ws_kboxes = ws_kvals  + (size_t)BATCH * NCLS * PMAX;      // B*C*P*4

    float* out_boxes  = (float*)d_out;
    float* out_vals   = out_boxes + (size_t)BATCH * MAXTOT * 4;
    float* out_labels = out_vals  + (size_t)BATCH * MAXTOT;

    const int tot = BATCH * NANCH;
    k_decode_mask<<<(tot + TPB - 1) / TPB, TPB, 0, stream>>>(
        deltas, labels, anchors, ws_boxes, ws_mask);
    k_topk<<<BATCH * NCLS, TPB, 0, stream>>>(
        labels, ws_mask, ws_boxes, ws_cvals, ws_cboxes);
    k_nms<<<BATCH * NCLS, TPB, 0, stream>>>(
        ws_cvals, ws_cboxes, ws_kvals, ws_kboxes);
    k_final<<<BATCH, FTPB, 0, stream>>>(
        ws_kvals, ws_kboxes, out_boxes, out_vals, out_labels);
}